// ShortConvolution_154618823095
// MI455X (gfx1250) — compile-verified
//
#include <hip/hip_runtime.h>
#include <cstdint>
#include <cstddef>

// Problem constants (match reference)
#define B_ 4
#define L_ 4096
#define H_ 2048
#define K_ 4
#define EPSF 1e-5f

// Tiling
#define TPB   256          // 8 waves of 32
#define CPT   8            // channels per thread: 256*8 = 2048 = H
#define CROWS 32           // L-rows per block
#define LA    3            // async prefetch lookahead (rows)
#define SLOTS 7            // LDS ring slots; need >= (K-1)+LA+1 = 7
#define WAVES (TPB / 32)

typedef float v4f __attribute__((ext_vector_type(4)));
typedef int   v4i __attribute__((ext_vector_type(4)));
typedef __attribute__((address_space(1))) float gbl_float;
typedef __attribute__((address_space(3))) float lds_float;
typedef __attribute__((address_space(1))) v4i   gbl_v4i;
typedef __attribute__((address_space(3))) v4i   lds_v4i;

#if defined(__has_builtin)
#if __has_builtin(__builtin_amdgcn_global_load_async_to_lds_b128)
#define ASYNC_PATH 1
#else
#define ASYNC_PATH 0
#endif
#else
#define ASYNC_PATH 0
#endif

__device__ __forceinline__ void row_load_16B(const float* gp, float* lp) {
#if ASYNC_PATH
  // async DMA: global -> LDS, 16B per lane, tracked by ASYNCcnt.
  // Builtin wants <4 x i32> pointers: addrspacecast, then pointee bitcast.
  gbl_v4i* gsrc = (gbl_v4i*)(gbl_float*)gp;
  lds_v4i* ldst = (lds_v4i*)(lds_float*)lp;
  __builtin_amdgcn_global_load_async_to_lds_b128(gsrc, ldst,
                                                 /*imm offset*/ 0, /*cpol*/ 0);
#else
  *(v4f*)lp = *(const v4f*)gp;
#endif
}

// one H-row: each thread moves 32B (2 x b128) -> 2 async ops per wave per row
__device__ __forceinline__ void issue_row(const float* xrow, float* lrow) {
  row_load_16B(xrow, lrow);
  row_load_16B(xrow + 4, lrow + 4);
}

template <int N>
__device__ __forceinline__ void wait_async_le() {
#if ASYNC_PATH
#if __has_builtin(__builtin_amdgcn_s_wait_asynccnt)
  __builtin_amdgcn_s_wait_asynccnt(N);
#else
  asm volatile("s_wait_asynccnt %0" ::"n"(N));
#endif
#endif
}

__device__ __forceinline__ int slot_of(int r) {
  // r >= l0-3 >= -3, so r + (K-1) >= 0
  return (int)((unsigned)(r + (K_ - 1)) % (unsigned)SLOTS);
}

__global__ __launch_bounds__(TPB) void ShortConvolution_154618823095_kernel(
    const float* __restrict__ x, const float* __restrict__ w,
    const float* __restrict__ g_, const float* __restrict__ b_,
    float* __restrict__ out) {
  __shared__ float ring[SLOTS][H_];  // 7 * 8KB = 56KB
  __shared__ float red1[WAVES];
  __shared__ float red2[WAVES];

  const int tid = threadIdx.x;
  const int c0 = tid * CPT;
  const int tiles_per_b = L_ / CROWS;
  const int b = blockIdx.x / tiles_per_b;
  const int l0 = (blockIdx.x % tiles_per_b) * CROWS;

  const float* xb = x + (size_t)b * L_ * H_;
  float* ob = out + (size_t)b * L_ * H_;

  // Per-thread constants: 4 taps + gamma + beta for 8 channels
  float wk[K_][CPT], gm[CPT], bt[CPT];
#pragma unroll
  for (int j = 0; j < CPT; ++j) {
    v4f t = *(const v4f*)(w + (size_t)(c0 + j) * K_);
    wk[0][j] = t.x; wk[1][j] = t.y; wk[2][j] = t.z; wk[3][j] = t.w;
  }
  {
    v4f t0 = *(const v4f*)(g_ + c0);
    v4f t1 = *(const v4f*)(g_ + c0 + 4);
    gm[0] = t0.x; gm[1] = t0.y; gm[2] = t0.z; gm[3] = t0.w;
    gm[4] = t1.x; gm[5] = t1.y; gm[6] = t1.z; gm[7] = t1.w;
    v4f u0 = *(const v4f*)(b_ + c0);
    v4f u1 = *(const v4f*)(b_ + c0 + 4);
    bt[0] = u0.x; bt[1] = u0.y; bt[2] = u0.z; bt[3] = u0.w;
    bt[4] = u1.x; bt[5] = u1.y; bt[6] = u1.z; bt[7] = u1.w;
  }

  // Prologue: halo rows l0-3..l0-1 (zero-fill if before sequence start)
  for (int r = l0 - (K_ - 1); r < l0; ++r) {
    float* lp = &ring[slot_of(r)][c0];
    if (r >= 0) {
      issue_row(xb + (size_t)r * H_ + c0, lp);
    } else {
      v4f z = {0.f, 0.f, 0.f, 0.f};
      *(v4f*)lp = z;
      *(v4f*)(lp + 4) = z;
    }
  }
  // Prime LA rows of the pipeline
#pragma unroll
  for (int i = 0; i < LA; ++i) {
    const int r = l0 + i;
    issue_row(xb + (size_t)r * H_ + c0, &ring[slot_of(r)][c0]);
  }

  for (int i = 0; i < CROWS; ++i) {
    const int r = l0 + i;
    // row r complete when <= 2*(LA-1) async ops remain in flight (in-order)
    if (i <= CROWS - LA) wait_async_le<2 * (LA - 1)>();
    else                 wait_async_le<0>();
    __syncthreads();  // all waves' DMA portions of row r visible in LDS

    // Fused causal depthwise conv + residual
    float y[CPT];
    {
      const float* cur = &ring[slot_of(r)][c0];
      v4f a = *(const v4f*)cur;
      v4f c = *(const v4f*)(cur + 4);
      float xc[CPT] = {a.x, a.y, a.z, a.w, c.x, c.y, c.z, c.w};
#pragma unroll
      for (int j = 0; j < CPT; ++j) y[j] = xc[j] + wk[K_ - 1][j] * xc[j];
#pragma unroll
      for (int k = 0; k < K_ - 1; ++k) {
        const float* tp = &ring[slot_of(r - (K_ - 1) + k)][c0];
        v4f u = *(const v4f*)tp;
        v4f v = *(const v4f*)(tp + 4);
        float xt[CPT] = {u.x, u.y, u.z, u.w, v.x, v.y, v.z, v.w};
#pragma unroll
        for (int j = 0; j < CPT; ++j) y[j] += wk[k][j] * xt[j];
      }
    }

    // Prefetch row r+LA into slot(r+LA)==slot(r-4): disjoint from live window r-3..r
    if (i + LA < CROWS) {
      const int rr = r + LA;
      issue_row(xb + (size_t)rr * H_ + c0, &ring[slot_of(rr)][c0]);
    }

    // LayerNorm over H: wave32 shuffle reduce, then 8-wave LDS reduce
    float s1 = 0.f, s2 = 0.f;
#pragma unroll
    for (int j = 0; j < CPT; ++j) { s1 += y[j]; s2 += y[j] * y[j]; }
#pragma unroll
    for (int off = 16; off > 0; off >>= 1) {
      s1 += __shfl_xor(s1, off, 32);
      s2 += __shfl_xor(s2, off, 32);
    }
    const int wv = tid >> 5;
    if ((tid & 31) == 0) { red1[wv] = s1; red2[wv] = s2; }
    __syncthreads();
    float t1 = 0.f, t2 = 0.f;
#pragma unroll
    for (int q = 0; q < WAVES; ++q) { t1 += red1[q]; t2 += red2[q]; }
    const float mean = t1 * (1.0f / (float)H_);
    const float var  = t2 * (1.0f / (float)H_) - mean * mean;
    const float rstd = rsqrtf(var + EPSF);

    float o[CPT];
#pragma unroll
    for (int j = 0; j < CPT; ++j) o[j] = (y[j] - mean) * rstd * gm[j] + bt[j];
    v4f o0 = {o[0], o[1], o[2], o[3]};
    v4f o1 = {o[4], o[5], o[6], o[7]};
    float* op = ob + (size_t)r * H_ + c0;
    __builtin_nontemporal_store(o0, (v4f*)op);        // TH=NT: output never re-read
    __builtin_nontemporal_store(o1, (v4f*)(op + 4));
  }
}

extern "C" void kernel_launch(void* const* d_in, const int* in_sizes, int n_in,
                              void* d_out, int out_size, void* d_ws, size_t ws_size,
                              hipStream_t stream) {
  (void)in_sizes; (void)n_in; (void)out_size; (void)d_ws; (void)ws_size;
  const float* x     = (const float*)d_in[0];
  const float* w     = (const float*)d_in[1];
  const float* gamma = (const float*)d_in[2];
  const float* beta  = (const float*)d_in[3];
  float* out = (float*)d_out;

  dim3 grid(B_ * (L_ / CROWS));  // 512 blocks
  dim3 block(TPB);
  ShortConvolution_154618823095_kernel<<<grid, block, 0, stream>>>(x, w, gamma, beta, out);
}